// PointConv_33852932227613
// MI455X (gfx1250) — compile-verified
//
#include <hip/hip_runtime.h>

typedef float v2f __attribute__((ext_vector_type(2)));
typedef float v8f __attribute__((ext_vector_type(8)));
typedef unsigned uv4 __attribute__((ext_vector_type(4)));
typedef int      iv4 __attribute__((ext_vector_type(4)));
typedef int      iv8 __attribute__((ext_vector_type(8)));

#define B_  32      // batch
#define C_  64      // channels
#define N_  256     // n = C*KH*KW  (GEMM M and K)
#define L_  1024    // spatial locations (nH*nW)
#define XS  258     // padded k-stride for Xs[b][k] (258 mod 64 = 2 -> conflict-free b64 reads)
#define KC  32      // K-rows per TDM chunk
#define AS  40      // LDS A-row stride in dwords: 32 data + 8 TDM pad (conflict-free)
#define NCH (N_ / KC)

#if __has_builtin(__builtin_amdgcn_tensor_load_to_lds) && \
    __has_builtin(__builtin_amdgcn_s_wait_tensorcnt)
#define USE_TDM 1
#else
#define USE_TDM 0
#endif

__device__ __forceinline__ v8f wmma4(v2f a, v2f b, v8f c) {
  // V_WMMA_F32_16X16X4_F32 : D(16x16,f32) = A(16x4,f32) * B(4x16,f32) + C
  return __builtin_amdgcn_wmma_f32_16x16x4_f32(
      /*neg_a=*/false, a, /*neg_b=*/false, b,
      /*c_mod=*/(short)0, c, /*reuse_a=*/false, /*reuse_b=*/false);
}

__device__ __forceinline__ void wait_async0() {
#if __has_builtin(__builtin_amdgcn_s_wait_asynccnt)
  __builtin_amdgcn_s_wait_asynccnt(0);
#else
  asm volatile("s_wait_asynccnt 0x0" ::: "memory");
#endif
}

#if USE_TDM
// Issue a TDM load of a 32x32 f32 tile (tile rows contiguous 32 floats,
// row stride 256 floats in memory) into LDS with 8-dword padding per row.
// D# bitfields per CDNA5 ISA 8.3/8.4. Groups 2/3 unused (2D tensor) -> zeros.
__device__ __forceinline__ void tdm_issue_w(const float* gaddr, unsigned lds_off) {
  const unsigned long long ga = (unsigned long long)(size_t)gaddr;
  uv4 g0;
  g0[0] = 1u;                                   // count=1, user descriptor
  g0[1] = lds_off;                              // lds_addr (bytes)
  g0[2] = (unsigned)ga;                         // global_addr[31:0]
  g0[3] = (unsigned)((ga >> 32) & 0x01FFFFFFu)  // global_addr[56:32]
          | 0x80000000u;                        // type=2 ("image")
  iv8 g1;
  g1[0] = (int)0x0F120000u;   // data_size=4B | pad_enable | pad_interval=32dw | pad_amount=8dw
  g1[1] = (int)(256u << 16);  // tensor_dim0 = 256  (bits 79:48, low part)
  g1[2] = (int)(256u << 16);  // tensor_dim1 = 256  (bits 111:80, low part)
  g1[3] = (int)(32u << 16);   // tile_dim0 = 32     (bits 127:112)
  g1[4] = 32;                 // tile_dim1 = 32     (bits 143:128); tile_dim2=0
  g1[5] = 256;                // tensor_dim0_stride = 256 (bits 207:160)
  g1[6] = 0;
  g1[7] = 0;
  const iv4 z4 = {0, 0, 0, 0};
  const iv8 z8 = {0, 0, 0, 0, 0, 0, 0, 0};
  // 6-arg form on this toolchain: (g0, g1, g2, g3, <aux>, cpol)
  __builtin_amdgcn_tensor_load_to_lds(g0, g1, z4, z4, z8, 0);
}
#endif

// Consume one KC-row A chunk from LDS against Xs, accumulating 2x2 WMMA tiles.
__device__ __forceinline__ void consume_chunk(const float* __restrict__ Ab,
                                              const float* __restrict__ Xsp,
                                              int kg0, int lh, int ll,
                                              v8f acc[2][2]) {
#pragma unroll 4
  for (int kq = 0; kq < KC; kq += 4) {
    const int kl = kq + (lh << 1);   // lanes 16-31 take K+2,K+3
    const int kg = kg0 + kl;
    v2f a0, a1;
    a0.x = Ab[ kl      * AS + ll];
    a0.y = Ab[(kl + 1) * AS + ll];
    a1.x = Ab[ kl      * AS + 16 + ll];
    a1.y = Ab[(kl + 1) * AS + 16 + ll];
    const v2f b0 = *(const v2f*)&Xsp[ ll       * XS + kg];
    const v2f b1 = *(const v2f*)&Xsp[(16 + ll) * XS + kg];
    acc[0][0] = wmma4(a0, b0, acc[0][0]);
    acc[0][1] = wmma4(a0, b1, acc[0][1]);
    acc[1][0] = wmma4(a1, b0, acc[1][0]);
    acc[1][1] = wmma4(a1, b1, acc[1][1]);
  }
}

__global__ __launch_bounds__(256)
void pointconv_wmma_f32(const float* __restrict__ x,
                        const float* __restrict__ weight,
                        const float* __restrict__ bias,
                        float* __restrict__ out) {
  __shared__ __align__(16) float Xs[B_ * XS];     // B tile [b][k]+pad (~33 KiB)
  __shared__ __align__(16) float Ds[N_ * B_];     // D tile [m][b]      (32 KiB)
#if USE_TDM
  __shared__ __align__(16) float As[16][KC * AS]; // 8 waves x 2 buffers (80 KiB)
#endif

  const int l   = blockIdx.x;      // l = ph*32 + pw
  const int ph  = l >> 5;
  const int pw  = l & 31;
  const int h0  = ph << 1;
  const int w0  = pw << 1;
  const int tid = threadIdx.x;

  const int wave = tid >> 5;
  const int lane = tid & 31;
  const int lh   = lane >> 4;      // K-half selector (A/B), M-half (C/D)
  const int ll   = lane & 15;      // M within A tile / N within B tile
  const int M0   = wave << 5;      // this wave owns m in [M0, M0+32)

  const float* __restrict__ Wl = weight + (size_t)l * (N_ * N_); // W_l[k][m]

#if USE_TDM
  // Kick off the first weight-slab DMA early (TDM, per-wave TENSORcnt).
  const float* Wlw = Wl + M0;      // 32-column slab owned by this wave
  const unsigned asbuf[2] = {(unsigned)(size_t)&As[(wave << 1) + 0][0],
                             (unsigned)(size_t)&As[(wave << 1) + 1][0]};
  tdm_issue_w(Wlw, asbuf[0]);
#endif

  // ---- unfold: async DMA X_l into LDS as Xs[b*XS + k], k = c*4 + kh*2 + kw ----
  for (int i = tid; i < B_ * C_ * 2; i += 256) {
    const int kh = i & 1;
    const int c  = (i >> 1) & 63;
    const int b  = i >> 7;
    const float* gsrc = x + ((((b << 6) + c) << 6) + h0 + kh) * 64 + w0;
    const unsigned lds_off =
        (unsigned)(size_t)&Xs[b * XS + (c << 2) + (kh << 1)];
    asm volatile("global_load_async_to_lds_b64 %0, %1, off"
                 :
                 : "v"(lds_off), "v"((unsigned long long)(size_t)gsrc)
                 : "memory");
  }
  wait_async0();
  __syncthreads();

  // ---- init accumulators with bias (bias[p*L + l]; same for both N tiles) ----
  v8f acc[2][2];
  for (int mt = 0; mt < 2; ++mt) {
    const int mb = M0 + (mt << 4) + (lh << 3);   // C-layout: lanes 16-31 hold M=8..15
    for (int vi = 0; vi < 8; ++vi) {
      const float bv = bias[(mb + vi) * L_ + l];
      acc[mt][0][vi] = bv;
      acc[mt][1][vi] = bv;
    }
  }

#if USE_TDM
  // ---- K loop: TDM double-buffered weight slabs, ds_load + wmma only ----
  for (int ci = 0; ci < NCH - 1; ++ci) {
    tdm_issue_w(Wlw + (size_t)(ci + 1) * (KC * N_), asbuf[(ci + 1) & 1]);
    __builtin_amdgcn_s_wait_tensorcnt((short)1);   // chunk ci landed (in-order)
    consume_chunk(&As[(wave << 1) + (ci & 1)][0], Xs, ci * KC, lh, ll, acc);
  }
  __builtin_amdgcn_s_wait_tensorcnt((short)0);
  consume_chunk(&As[(wave << 1) + ((NCH - 1) & 1)][0], Xs, (NCH - 1) * KC,
                lh, ll, acc);
#else
  // ---- fallback: stream A straight from global ----
#pragma unroll 4
  for (int k0 = 0; k0 < N_; k0 += 4) {
    const int ka = k0 + (lh << 1);
    if (k0 + 32 < N_) {
      __builtin_prefetch(&Wl[(ka + 32) * N_ + M0 + ll], 0, 3);
      __builtin_prefetch(&Wl[(ka + 32) * N_ + M0 + 16 + ll], 0, 3);
    }
    v2f a0, a1;
    a0.x = Wl[ ka      * N_ + M0 + ll];
    a0.y = Wl[(ka + 1) * N_ + M0 + ll];
    a1.x = Wl[ ka      * N_ + M0 + 16 + ll];
    a1.y = Wl[(ka + 1) * N_ + M0 + 16 + ll];
    const v2f b0 = *(const v2f*)&Xs[ ll       * XS + ka];
    const v2f b1 = *(const v2f*)&Xs[(16 + ll) * XS + ka];
    acc[0][0] = wmma4(a0, b0, acc[0][0]);
    acc[0][1] = wmma4(a0, b1, acc[0][1]);
    acc[1][0] = wmma4(a1, b0, acc[1][0]);
    acc[1][1] = wmma4(a1, b1, acc[1][1]);
  }
#endif

  // ---- write accumulators to LDS in [m][b] order ----
  for (int mt = 0; mt < 2; ++mt)
    for (int nt = 0; nt < 2; ++nt)
      for (int vi = 0; vi < 8; ++vi)
        Ds[(M0 + (mt << 4) + (lh << 3) + vi) * B_ + (nt << 4) + ll] =
            acc[mt][nt][vi];
  __syncthreads();

  // ---- fold: LDS -> NCHW with aligned float2 stores ----
  for (int i = tid; i < B_ * C_ * 2; i += 256) {
    const int kh = i & 1;
    const int c  = (i >> 1) & 63;
    const int b  = i >> 7;
    const int k  = (c << 2) + (kh << 1);
    float2 v;
    v.x = Ds[ k      * B_ + b];
    v.y = Ds[(k + 1) * B_ + b];
    *(float2*)(out + ((((b << 6) + c) << 6) + h0 + kh) * 64 + w0) = v;
  }
}

extern "C" void kernel_launch(void* const* d_in, const int* in_sizes, int n_in,
                              void* d_out, int out_size, void* d_ws, size_t ws_size,
                              hipStream_t stream) {
  const float* x      = (const float*)d_in[0];  // (32,64,64,64) f32
  const float* weight = (const float*)d_in[1];  // (1024,256,256) f32
  const float* bias   = (const float*)d_in[2];  // (1,1,256,1024) f32
  float* out          = (float*)d_out;          // (32,64,64,64) f32
  pointconv_wmma_f32<<<dim3(L_), dim3(256), 0, stream>>>(x, weight, bias, out);
}